// AdjacencyNet_13108240187968
// MI455X (gfx1250) — compile-verified
//
#include <hip/hip_runtime.h>
#include <hip/hip_bf16.h>

typedef float v2f __attribute__((ext_vector_type(2)));
typedef float v8f __attribute__((ext_vector_type(8)));

#define NB 8
#define NN 2048
#define ND 256
#define ADJ_ELEMS (NB * NN * NN)   // 33554432

// ws layout (float index):
//  [0..7]   per-branch sum of squares (float, atomicAdd)
//  [8..15]  per-branch max of raw adj (int bits, atomicMax; values >= 0)
//  [16]     loss accumulator
//  [24..31] per-branch 1/frob_norm
//  [32]     threshold (0.5 * global max of normalized adj)
//  [64..64+16383]      per-row 1/||x|| (16384 rows)
//  int[16448..24639]   chunk counts -> exclusive offsets (8192)
#define OFF_BMAX   8
#define OFF_LOSS   16
#define OFF_INV    24
#define OFF_THRESH 32
#define OFF_ROWINV 64
#define OFF_COUNTS 16448

__global__ void init_ws(float* ws) {
    if (threadIdx.x < 64) ws[threadIdx.x] = 0.0f;
}

__global__ __launch_bounds__(256) void rownorm_kernel(const float* __restrict__ X,
                                                      float* __restrict__ rowinv) {
    int row  = blockIdx.x * 8 + (threadIdx.x >> 5);
    int lane = threadIdx.x & 31;
    const float* p = X + (size_t)row * ND;
    float ss = 0.0f;
    for (int d = lane; d < ND; d += 32) { float v = p[d]; ss += v * v; }
    for (int m = 16; m > 0; m >>= 1) ss += __shfl_xor(ss, m, 32);
    if (lane == 0) rowinv[row] = 1.0f / sqrtf(ss);
}

// MODE 0: adj_raw = relu(X_b @ W^T + bias); accumulate per-branch sumsq & max.
// MODE 1: sim = (X_b @ X_b^T) scaled by rowinv; loss += (adj_norm - sim)^2;
//         adj <- threshold(adj_norm) in place.
template <int MODE>
__global__ __launch_bounds__(256) void gemm_wmma(const float* __restrict__ X,
                                                 const float* __restrict__ Bsrc,
                                                 const float* __restrict__ bias,
                                                 float* __restrict__ adj,
                                                 float* __restrict__ ws) {
    __shared__ float Al[128][33];
    __shared__ float Bl[128][33];

    const int tid  = threadIdx.x;
    const int wave = tid >> 5;
    const int lane = tid & 31;
    const int hi   = lane >> 4;
    const int ln   = lane & 15;
    const int bc   = blockIdx.z;
    const int R0   = blockIdx.y * 128;
    const int C0   = blockIdx.x * 128;

    const float* Abase = X + ((size_t)bc * NN + R0) * ND;
    const float* Bbase = (MODE == 0) ? (Bsrc + (size_t)C0 * ND)
                                     : (Bsrc + ((size_t)bc * NN + C0) * ND);

    v8f acc[8];
    for (int i = 0; i < 8; i++)
        for (int j = 0; j < 8; j++) acc[i][j] = 0.0f;

    const int tr  = tid >> 3;        // 0..31
    const int tc4 = (tid & 7) * 4;   // 0..28 step 4
    const int arow = wave * 16 + ln;

    for (int kk = 0; kk < ND; kk += 32) {
        __syncthreads();
        for (int i = 0; i < 4; i++) {
            int row = tr + 32 * i;
            const float4 av = *(const float4*)(Abase + (size_t)row * ND + kk + tc4);
            Al[row][tc4 + 0] = av.x; Al[row][tc4 + 1] = av.y;
            Al[row][tc4 + 2] = av.z; Al[row][tc4 + 3] = av.w;
            const float4 bv = *(const float4*)(Bbase + (size_t)row * ND + kk + tc4);
            Bl[row][tc4 + 0] = bv.x; Bl[row][tc4 + 1] = bv.y;
            Bl[row][tc4 + 2] = bv.z; Bl[row][tc4 + 3] = bv.w;
        }
        __syncthreads();
        for (int ks = 0; ks < 8; ks++) {
            int kb = ks * 4 + 2 * hi;                 // K = 2*hi + v, v in {0,1}
            v2f af; af[0] = Al[arow][kb]; af[1] = Al[arow][kb + 1];
            for (int ct = 0; ct < 8; ct++) {
                v2f bf; bf[0] = Bl[ct * 16 + ln][kb]; bf[1] = Bl[ct * 16 + ln][kb + 1];
                acc[ct] = __builtin_amdgcn_wmma_f32_16x16x4_f32(
                    false, af, false, bf, (short)0, acc[ct], false, false);
            }
        }
    }

    const int rowb = R0 + wave * 16 + hi * 8;   // D layout: M = r + 8*hi
    float lsum = 0.0f, lmax = 0.0f;

    if (MODE == 0) {
        for (int ct = 0; ct < 8; ct++) {
            int col = C0 + ct * 16 + ln;
            float bv = bias[col];
            for (int r = 0; r < 8; r++) {
                float v = fmaxf(acc[ct][r] + bv, 0.0f);
                adj[((size_t)bc << 22) + (size_t)(rowb + r) * NN + col] = v;
                lsum += v * v;
                lmax = fmaxf(lmax, v);
            }
        }
    } else {
        const float  invb = ws[OFF_INV + bc];
        const float  thr  = ws[OFF_THRESH];
        const float* rinv = ws + OFF_ROWINV;
        for (int ct = 0; ct < 8; ct++) {
            int col = C0 + ct * 16 + ln;
            float civ = rinv[bc * NN + col];
            for (int r = 0; r < 8; r++) {
                float sim = acc[ct][r] * rinv[bc * NN + rowb + r] * civ;
                size_t off = ((size_t)bc << 22) + (size_t)(rowb + r) * NN + col;
                float a = adj[off] * invb;
                float d = a - sim;
                lsum += d * d;
                adj[off] = (a < thr) ? 0.0f : a;
            }
        }
    }

    __syncthreads();
    float* redS = &Al[0][0];
    float* redM = &Bl[0][0];
    redS[tid] = lsum;
    if (MODE == 0) redM[tid] = lmax;
    __syncthreads();
    for (int s = 128; s > 0; s >>= 1) {
        if (tid < s) {
            redS[tid] += redS[tid + s];
            if (MODE == 0) redM[tid] = fmaxf(redM[tid], redM[tid + s]);
        }
        __syncthreads();
    }
    if (tid == 0) {
        if (MODE == 0) {
            atomicAdd(&ws[bc], redS[0]);
            atomicMax((int*)ws + OFF_BMAX + bc, __float_as_int(redM[0]));
        } else {
            atomicAdd(&ws[OFF_LOSS], redS[0]);
        }
    }
}

__global__ void finalize_kernel(float* ws) {
    int t = threadIdx.x;
    __shared__ float inv[8];
    if (t < 8) {
        float iv = 1.0f / sqrtf(ws[t]);
        ws[OFF_INV + t] = iv;
        inv[t] = iv;
    }
    __syncthreads();
    if (t == 0) {
        const int* mi = (const int*)ws;
        float g = 0.0f;
        for (int b = 0; b < 8; b++)
            g = fmaxf(g, __int_as_float(mi[OFF_BMAX + b]) * inv[b]);
        ws[OFF_THRESH] = 0.5f * g;
    }
}

__global__ __launch_bounds__(256) void count_kernel(const float* __restrict__ adj,
                                                    int* __restrict__ counts) {
    __shared__ int red[256];
    int t = threadIdx.x;
    size_t base = (size_t)blockIdx.x * 4096;
    int c = 0;
    for (int i = 0; i < 16; i++) c += (adj[base + t + i * 256] != 0.0f) ? 1 : 0;
    red[t] = c;
    __syncthreads();
    for (int s = 128; s > 0; s >>= 1) { if (t < s) red[t] += red[t + s]; __syncthreads(); }
    if (t == 0) counts[blockIdx.x] = red[0];
}

__global__ __launch_bounds__(256) void scan_kernel(int* counts) {
    __shared__ int part[256];
    int t = threadIdx.x;
    int base = t * 32;
    int s = 0;
    for (int i = 0; i < 32; i++) s += counts[base + i];
    part[t] = s;
    __syncthreads();
    for (int off = 1; off < 256; off <<= 1) {
        int v = part[t];
        int add = (t >= off) ? part[t - off] : 0;
        __syncthreads();
        part[t] = v + add;
        __syncthreads();
    }
    int run = (t == 0) ? 0 : part[t - 1];
    for (int i = 0; i < 32; i++) {
        int tmp = counts[base + i];
        counts[base + i] = run;
        run += tmp;
    }
}

__global__ void scatter_kernel(const float* __restrict__ adj,
                               const int* __restrict__ offs,
                               float* __restrict__ edges, int nnz) {
    int lane = threadIdx.x;
    size_t base = (size_t)blockIdx.x * 4096;
    int pos = offs[blockIdx.x];
    for (int it = 0; it < 128; it++) {
        size_t idx = base + (size_t)it * 32 + lane;
        float v = adj[idx];
        bool p = (v != 0.0f);
        unsigned int m = (unsigned int)__ballot(p);
        int pre = __popc(m & ((1u << lane) - 1u));
        if (p) {
            int w = pos + pre;
            if (w < nnz) {
                int b = (int)(idx >> 22);
                int r = (int)((idx >> 11) & 2047);
                int c = (int)(idx & 2047);
                edges[w]       = (float)(b * NN + r);
                edges[nnz + w] = (float)(b * NN + c);
            }
        }
        pos += __popc(m);
    }
}

__global__ void loss_kernel(const float* ws, float* out, int out_size) {
    out[out_size - 1] = ws[OFF_LOSS] * 0.125f;   // mean over 8 branches
}

extern "C" void kernel_launch(void* const* d_in, const int* in_sizes, int n_in,
                              void* d_out, int out_size, void* d_ws, size_t ws_size,
                              hipStream_t stream) {
    const float* X    = (const float*)d_in[0];
    const float* W    = (const float*)d_in[1];
    const float* bias = (const float*)d_in[2];
    float* out = (float*)d_out;
    float* ws  = (float*)d_ws;
    int*   wsi = (int*)d_ws;

    int nnz = (out_size - ADJ_ELEMS - 1) / 2;

    init_ws<<<1, 64, 0, stream>>>(ws);
    rownorm_kernel<<<NB * NN / 8, 256, 0, stream>>>(X, ws + OFF_ROWINV);

    dim3 grid(NN / 128, NN / 128, NB);
    gemm_wmma<0><<<grid, 256, 0, stream>>>(X, W, bias, out, ws);
    finalize_kernel<<<1, 32, 0, stream>>>(ws);
    gemm_wmma<1><<<grid, 256, 0, stream>>>(X, X, bias, out, ws);

    count_kernel<<<ADJ_ELEMS / 4096, 256, 0, stream>>>(out, wsi + OFF_COUNTS);
    scan_kernel<<<1, 256, 0, stream>>>(wsi + OFF_COUNTS);
    scatter_kernel<<<ADJ_ELEMS / 4096, 32, 0, stream>>>(out, wsi + OFF_COUNTS,
                                                        out + ADJ_ELEMS, nnz);
    loss_kernel<<<1, 1, 0, stream>>>(ws, out, out_size);
}